// LGNDICE_89361089560758
// MI455X (gfx1250) — compile-verified
//
#include <hip/hip_runtime.h>
#include <math.h>

#define N_USER  100000
#define N_ITEM  50000
#define N_NODES 150000
#define DIMS    64
#define N_EDGES 2000000
#define BATCH   8192

typedef float v2f __attribute__((ext_vector_type(2)));
typedef float v8f __attribute__((ext_vector_type(8)));

// ---------------- utility kernels ----------------
__global__ void zero_f(float* __restrict__ p, int n) {
    int i = blockIdx.x * blockDim.x + threadIdx.x;
    if (i < n) p[i] = 0.0f;
}
__global__ void zero_i(int* __restrict__ p, int n) {
    int i = blockIdx.x * blockDim.x + threadIdx.x;
    if (i < n) p[i] = 0;
}
__global__ void scale_f4(float* __restrict__ p, int n4, float s) {
    int i = blockIdx.x * blockDim.x + threadIdx.x;
    if (i >= n4) return;
    float4 v = ((float4*)p)[i];
    v.x *= s; v.y *= s; v.z *= s; v.w *= s;
    ((float4*)p)[i] = v;
}

// ---------------- degree / norm ----------------
__global__ void degree_k(const int* __restrict__ dst, float* __restrict__ degs) {
    int e = blockIdx.x * blockDim.x + threadIdx.x;
    if (e < N_EDGES) atomicAdd(&degs[dst[e]], 1.0f);
}
__global__ void norm_k(const float* __restrict__ degs, float* __restrict__ nrm) {
    int n = blockIdx.x * blockDim.x + threadIdx.x;
    if (n < N_NODES) {
        float d = degs[n];
        d = d < 1.0f ? 1.0f : d;
        nrm[n] = 1.0f / sqrtf(d);
    }
}

// ---------------- LightGCN propagation ----------------
__global__ void init_feat(const float* __restrict__ emb, float* __restrict__ acc,
                          float* __restrict__ h, int n4) {
    int i = blockIdx.x * blockDim.x + threadIdx.x;
    if (i >= n4) return;
    float4 v = ((const float4*)emb)[i];
    ((float4*)acc)[i] = v;
    ((float4*)h)[i]   = v;
}

// 16 threads per edge, each handles 4 dims (float4): coalesced 256B per edge row.
__global__ void scatter_k(const float* __restrict__ h, const float* __restrict__ nrm,
                          const int* __restrict__ src, const int* __restrict__ dst,
                          float* __restrict__ nxt) {
    long long tid = (long long)blockIdx.x * blockDim.x + threadIdx.x;
    if (tid >= (long long)N_EDGES * 16) return;
    int e = (int)(tid >> 4);
    int g = (int)(tid & 15);
    int s = src[e];
    int d = dst[e];
    float ns = nrm[s];
    float4 v = ((const float4*)(h + (size_t)s * DIMS))[g];
    float* o = nxt + (size_t)d * DIMS + g * 4;
    atomicAdd(o + 0, v.x * ns);
    atomicAdd(o + 1, v.y * ns);
    atomicAdd(o + 2, v.z * ns);
    atomicAdd(o + 3, v.w * ns);
}

// h = nxt * norm[dst]; acc += h
__global__ void update_k(const float* __restrict__ nxt, const float* __restrict__ nrm,
                         float* __restrict__ h, float* __restrict__ acc, int n4) {
    int i = blockIdx.x * blockDim.x + threadIdx.x;
    if (i >= n4) return;
    float nd = nrm[i >> 4];               // 16 float4 per node
    float4 v = ((const float4*)nxt)[i];
    v.x *= nd; v.y *= nd; v.z *= nd; v.w *= nd;
    ((float4*)h)[i] = v;
    float4 a = ((float4*)acc)[i];
    a.x += v.x; a.y += v.y; a.z += v.z; a.w += v.w;
    ((float4*)acc)[i] = a;
}

// ---------------- unique flags ----------------
__global__ void set_flags(const int* __restrict__ user, const int* __restrict__ ip,
                          const int* __restrict__ in_, int* __restrict__ flags) {
    int i = blockIdx.x * blockDim.x + threadIdx.x;
    if (i >= BATCH) return;
    flags[user[i]] = 1;                 // users: indices < N_USER
    flags[ip[i] + N_USER] = 1;          // items: indices >= N_USER (disjoint range)
    flags[in_[i] + N_USER] = 1;
}

// ---------------- WMMA batched dot products ----------------
// One wave per 16-sample tile. D = U·P^T via V_WMMA_F32_16X16X4_F32 (16 K-chunks);
// diagonal of the 16x16 result is the per-sample dot product.
// A 16x4 f32 layout: lane L holds A[L%16][4kc+2*(L>=16)+{0,1}] in 2 VGPRs.
// B 4x16 f32 layout is symmetric -> identical addressing on P rows.
__global__ __launch_bounds__(32) void scores_wmma(
        const float* __restrict__ fi, const float* __restrict__ fp,
        const int* __restrict__ user, const int* __restrict__ ip, const int* __restrict__ in_,
        float* __restrict__ sp_i, float* __restrict__ sn_i,
        float* __restrict__ sp_p, float* __restrict__ sn_p) {
    int lane = threadIdx.x;
    int t = blockIdx.x;
    int l16 = lane & 15;
    int hi  = lane >> 4;                // 0 or 1
    int s   = t * 16 + l16;
    size_t u  = (size_t)user[s] * DIMS;
    size_t pi = (size_t)(ip[s] + N_USER) * DIMS;
    size_t ni = (size_t)(in_[s] + N_USER) * DIMS;

    v8f c_pi = {}; v8f c_ni = {}; v8f c_pp = {}; v8f c_np = {};
#pragma unroll
    for (int kc = 0; kc < 16; ++kc) {
        int off = kc * 4 + hi * 2;
        v2f a_i  = *(const v2f*)(fi + u  + off);
        v2f bp_i = *(const v2f*)(fi + pi + off);
        v2f bn_i = *(const v2f*)(fi + ni + off);
        v2f a_p  = *(const v2f*)(fp + u  + off);
        v2f bp_p = *(const v2f*)(fp + pi + off);
        v2f bn_p = *(const v2f*)(fp + ni + off);
        c_pi = __builtin_amdgcn_wmma_f32_16x16x4_f32(false, a_i, false, bp_i, (short)0, c_pi, false, false);
        c_ni = __builtin_amdgcn_wmma_f32_16x16x4_f32(false, a_i, false, bn_i, (short)0, c_ni, false, false);
        c_pp = __builtin_amdgcn_wmma_f32_16x16x4_f32(false, a_p, false, bp_p, (short)0, c_pp, false, false);
        c_np = __builtin_amdgcn_wmma_f32_16x16x4_f32(false, a_p, false, bn_p, (short)0, c_np, false, false);
    }
    // C/D layout: VGPR r, lanes 0-15 -> (M=r, N=lane); lanes 16-31 -> (M=r+8, N=lane-16).
    // Diagonal M==N: lane r holds it in c[r]; lane r+24 holds M=r+8 diagonal in c[r].
#pragma unroll
    for (int r = 0; r < 8; ++r) {
        if (lane == r) {
            int o = t * 16 + r;
            sp_i[o] = c_pi[r]; sn_i[o] = c_ni[r]; sp_p[o] = c_pp[r]; sn_p[o] = c_np[r];
        }
        if (lane == r + 24) {
            int o = t * 16 + r + 8;
            sp_i[o] = c_pi[r]; sn_i[o] = c_ni[r]; sp_p[o] = c_pp[r]; sn_p[o] = c_np[r];
        }
    }
}

// ---------------- loss reductions ----------------
__device__ __forceinline__ float logsig(float x) {
    return fminf(x, 0.0f) - log1pf(expf(-fabsf(x)));
}

__global__ void loss_red(const float* __restrict__ sp_i, const float* __restrict__ sn_i,
                         const float* __restrict__ sp_p, const float* __restrict__ sn_p,
                         const unsigned char* __restrict__ mask, float* __restrict__ red) {
    int i = blockIdx.x * blockDim.x + threadIdx.x;
    if (i >= BATCH) return;
    float psi = sp_i[i], nsi = sn_i[i], psp = sp_p[i], nsp = sn_p[i];
    float m = mask[i] ? 1.0f : 0.0f;
    float li  = m * logsig(psi - nsi);
    float lp1 = m * logsig(nsp - psp);
    float lp2 = (1.0f - m) * logsig(psp - nsp);
    float lt  = logsig((psi + psp) - (nsi + nsp));
    atomicAdd(&red[0], li);
    atomicAdd(&red[1], lp1);
    atomicAdd(&red[2], lp2);
    atomicAdd(&red[3], lt);
}

__global__ void disc_red(const float* __restrict__ fi, const float* __restrict__ fp,
                         const int* __restrict__ flags, float* __restrict__ red) {
    int n = blockIdx.x * blockDim.x + threadIdx.x;
    if (n >= N_NODES) return;
    if (!flags[n]) return;
    const float4* a = (const float4*)(fi + (size_t)n * DIMS);
    const float4* b = (const float4*)(fp + (size_t)n * DIMS);
    float s = 0.0f;
#pragma unroll
    for (int j = 0; j < 16; ++j) {
        float4 x = a[j], y = b[j];
        float dx = x.x - y.x, dy = x.y - y.y, dz = x.z - y.z, dw = x.w - y.w;
        s += dx * dx + dy * dy + dz * dz + dw * dw;
    }
    if (n < N_USER) { atomicAdd(&red[4], s); atomicAdd(&red[6], 1.0f); }
    else           { atomicAdd(&red[5], s); atomicAdd(&red[7], 1.0f); }
}

__global__ void finalize_k(const float* __restrict__ red, float* __restrict__ out) {
    if (blockIdx.x == 0 && threadIdx.x == 0) {
        const float invB = 1.0f / (float)BATCH;
        float loss_int   = -red[0] * invB;
        float loss_pop   = -(red[1] * invB) - (red[2] * invB);
        float loss_total = -red[3] * invB;
        float disc = red[5] / (red[7] * (float)DIMS) + red[4] / (red[6] * (float)DIMS);
        out[0] = 0.1f * loss_int + 0.1f * loss_pop + loss_total - 0.01f * disc;
    }
}

// ---------------- launch ----------------
extern "C" void kernel_launch(void* const* d_in, const int* in_sizes, int n_in,
                              void* d_out, int out_size, void* d_ws, size_t ws_size,
                              hipStream_t stream) {
    const int* user   = (const int*)d_in[0];
    const int* item_p = (const int*)d_in[1];
    const int* item_n = (const int*)d_in[2];
    const unsigned char* mask = (const unsigned char*)d_in[3];   // jax bool = 1 byte
    const int* esrc   = (const int*)d_in[4];
    const int* edst   = (const int*)d_in[5];
    const float* emb_int = (const float*)d_in[6];
    const float* emb_pop = (const float*)d_in[7];
    float* out = (float*)d_out;

    const size_t ND = (size_t)N_NODES * DIMS;     // 9.6M floats
    char* p = (char*)d_ws;
    float* acc_int = (float*)p; p += ND * 4;
    float* acc_pop = (float*)p; p += ND * 4;
    float* hbuf    = (float*)p; p += ND * 4;
    float* nxt     = (float*)p; p += ND * 4;
    float* degs    = (float*)p; p += (size_t)N_NODES * 4;
    float* nrm     = (float*)p; p += (size_t)N_NODES * 4;
    int*   flags   = (int*)p;   p += (size_t)N_NODES * 4;
    float* sp_i    = (float*)p; p += (size_t)BATCH * 4;
    float* sn_i    = (float*)p; p += (size_t)BATCH * 4;
    float* sp_p    = (float*)p; p += (size_t)BATCH * 4;
    float* sn_p    = (float*)p; p += (size_t)BATCH * 4;
    float* red     = (float*)p; p += 8 * 4;

    const int TB = 256;
    const int ndi  = (int)ND;
    const int n4   = N_NODES * (DIMS / 4);
    const int gN   = (N_NODES + TB - 1) / TB;
    const int gND  = (ndi + TB - 1) / TB;
    const int gN4  = (n4 + TB - 1) / TB;
    const int gE   = (N_EDGES + TB - 1) / TB;
    const int gE16 = (int)(((long long)N_EDGES * 16 + TB - 1) / TB);
    const int gB   = (BATCH + TB - 1) / TB;

    // init: degs, flags, reduction accumulators
    zero_f<<<gN, TB, 0, stream>>>(degs, N_NODES);
    zero_i<<<gN, TB, 0, stream>>>(flags, N_NODES);
    zero_f<<<1, 32, 0, stream>>>(red, 8);

    // degrees -> norm
    degree_k<<<gE, TB, 0, stream>>>(edst, degs);
    norm_k<<<gN, TB, 0, stream>>>(degs, nrm);

    // LightGCN propagation, tables processed sequentially (buffers stay L2-resident)
    const float* embs[2] = {emb_int, emb_pop};
    float* accs[2] = {acc_int, acc_pop};
    for (int t = 0; t < 2; ++t) {
        init_feat<<<gN4, TB, 0, stream>>>(embs[t], accs[t], hbuf, n4);
        for (int L = 0; L < 3; ++L) {
            zero_f<<<gND, TB, 0, stream>>>(nxt, ndi);
            scatter_k<<<gE16, TB, 0, stream>>>(hbuf, nrm, esrc, edst, nxt);
            update_k<<<gN4, TB, 0, stream>>>(nxt, nrm, hbuf, accs[t], n4);
        }
        scale_f4<<<gN4, TB, 0, stream>>>(accs[t], n4, 0.25f);
    }

    // unique-row flags for discrepancy
    set_flags<<<gB, TB, 0, stream>>>(user, item_p, item_n, flags);

    // batched dot products via WMMA f32 16x16x4 (one wave per 16-sample tile)
    scores_wmma<<<BATCH / 16, 32, 0, stream>>>(acc_int, acc_pop, user, item_p, item_n,
                                               sp_i, sn_i, sp_p, sn_p);

    // reductions + final scalar
    loss_red<<<gB, TB, 0, stream>>>(sp_i, sn_i, sp_p, sn_p, mask, red);
    disc_red<<<gN, TB, 0, stream>>>(acc_int, acc_pop, flags, red);
    finalize_k<<<1, 1, 0, stream>>>(red, out);
}